// ForwardCASSI_38439957299667
// MI455X (gfx1250) — compile-verified
//
#include <hip/hip_runtime.h>
#include <stddef.h>
#include <stdint.h>

#define AS1 __attribute__((address_space(1)))
#define AS3 __attribute__((address_space(3)))

// Match the builtin's parameter type exactly: 'int __vector(4)' (gcc vector_size style)
typedef int v4i __attribute__((vector_size(4 * sizeof(int))));

// Problem shape (compile-time, from the reference)
constexpr int B_ = 8;
constexpr int M_ = 512;                 // spatial
constexpr int L_ = 31;                  // spectral bands
constexpr int NL = M_ * L_;             // 15872 floats per (b,m) slab = 63488 B
constexpr int YW = M_ + L_ - 1;         // 542 output columns
constexpr int Y_ELEMS = B_ * M_ * YW;   // 2220032
constexpr int H_ELEMS = M_ * M_;        // 262144 (second tuple output, passthrough)

// ---------------------------------------------------------------------------
// CDNA5 async global->LDS copy (ASYNCcnt-tracked path)
// ---------------------------------------------------------------------------
__device__ __forceinline__ void async_copy_b128(void* lds_dst, const void* gsrc) {
#if __has_builtin(__builtin_amdgcn_global_load_async_to_lds_b128)
  __builtin_amdgcn_global_load_async_to_lds_b128(
      (AS1 v4i*)gsrc, (AS3 v4i*)lds_dst, /*imm offset*/0, /*cpol*/0);
#else
  unsigned lds_off = (unsigned)(size_t)(AS3 void*)lds_dst;        // 32-bit LDS byte addr
  unsigned long long gaddr = (unsigned long long)(uintptr_t)gsrc; // 64-bit global addr
  asm volatile("global_load_async_to_lds_b128 %0, %1, off"
               :: "v"(lds_off), "v"(gaddr)
               : "memory");
#endif
}

__device__ __forceinline__ void wait_async_zero() {
#if __has_builtin(__builtin_amdgcn_s_wait_asynccnt)
  __builtin_amdgcn_s_wait_asynccnt(0);
#else
  asm volatile("s_wait_asynccnt 0x0" ::: "memory");
#endif
}

// ---------------------------------------------------------------------------
// Kernel 0: zero the global-max cell in the workspace (d_ws is poisoned once
// and never re-poisoned, so we must re-init every call for determinism).
// ---------------------------------------------------------------------------
__global__ void cassi_init_max(unsigned* gmax) { gmax[0] = 0u; }

// ---------------------------------------------------------------------------
// Kernel 1: one workgroup per (b,m) row.
//   - async-stage x[b,m,:,:] (63488 B) and H[m,:] (2048 B) into LDS
//   - each thread gathers the 31-tap diagonal per output column from LDS
//   - block max -> atomicMax on float bit pattern (all values >= 0)
// ---------------------------------------------------------------------------
__global__ __launch_bounds__(256) void cassi_row_kernel(
    const float* __restrict__ x, const float* __restrict__ H,
    float* __restrict__ Y, unsigned* __restrict__ gmax) {
  __shared__ float xs[NL];   // 63488 B, x slab for this row
  __shared__ float hs[M_];   // 2048 B, mask row
  __shared__ float red[256]; // max reduction

  const int row = blockIdx.x;       // row = b*M_ + m
  const int m   = row & (M_ - 1);
  const int tid = threadIdx.x;

  const float* gx = x + (size_t)row * NL;   // 16B-aligned: 63488 % 16 == 0
  const float* gh = H + (size_t)m * M_;     // 2048B-aligned

  // Stage x slab: 3968 x b128 chunks across 256 lanes (16 iterations, tail-guarded)
  for (int i = tid; i < NL / 4; i += 256) {
    async_copy_b128(&xs[i * 4], gx + i * 4);
  }
  // Stage H row: 128 x b128 chunks
  if (tid < M_ / 4) {
    async_copy_b128(&hs[tid * 4], gh + tid * 4);
  }

  wait_async_zero();   // this wave's async transfers done (LDS written)
  __syncthreads();     // all waves' transfers visible workgroup-wide

  // Diagonal gather: Y[c] = sum_l hs[c-l] * xs[(c-l)*31 + l]
  // Adjacent threads read LDS with stride 31 floats: conflict-free over 64 banks
  // (gcd(31,64)==1).
  float lmax = 0.0f;
  for (int c = tid; c < YW; c += 256) {
    int l0 = c - (M_ - 1); if (l0 < 0) l0 = 0;
    int l1 = (c < L_ - 1) ? c : (L_ - 1);
    float acc = 0.0f;
    #pragma unroll 4
    for (int l = l0; l <= l1; ++l) {
      const int n = c - l;
      acc = fmaf(hs[n], xs[n * L_ + l], acc);
    }
    Y[(size_t)row * YW + c] = acc;
    lmax = fmaxf(lmax, acc);
  }

  // Block max reduction (values >= 0, so uint bit-pattern max == float max,
  // and max is order-independent => deterministic across replays)
  red[tid] = lmax;
  __syncthreads();
  for (int s = 128; s > 0; s >>= 1) {
    if (tid < s) red[tid] = fmaxf(red[tid], red[tid + s]);
    __syncthreads();
  }
  if (tid == 0) atomicMax(gmax, __float_as_uint(red[0]));
}

// ---------------------------------------------------------------------------
// Kernel 2: normalize Y in place; append passthrough H (tuple output #2).
// ---------------------------------------------------------------------------
__global__ __launch_bounds__(256) void cassi_finalize(
    float* __restrict__ out, const float* __restrict__ H,
    const unsigned* __restrict__ gmax, int y_elems, int total) {
  int i = blockIdx.x * blockDim.x + threadIdx.x;
  if (i >= total) return;
  if (i < y_elems) {
    const float mv = __uint_as_float(*gmax);
    out[i] = out[i] / mv;             // match reference: true division by max
  } else {
    out[i] = H[i - y_elems];          // H passthrough
  }
}

// ---------------------------------------------------------------------------
extern "C" void kernel_launch(void* const* d_in, const int* in_sizes, int n_in,
                              void* d_out, int out_size, void* d_ws, size_t ws_size,
                              hipStream_t stream) {
  const float* x = (const float*)d_in[0];  // (8,512,512,31,1)
  const float* H = (const float*)d_in[1];  // (1,512,512,1,1)
  float* out = (float*)d_out;              // Y (8,512,542,1) ++ H flat
  unsigned* gmax = (unsigned*)d_ws;

  cassi_init_max<<<1, 1, 0, stream>>>(gmax);
  cassi_row_kernel<<<B_ * M_, 256, 0, stream>>>(x, H, out, gmax);

  const int total = Y_ELEMS + H_ELEMS;
  cassi_finalize<<<(total + 255) / 256, 256, 0, stream>>>(out, H, gmax, Y_ELEMS, total);
}